// vSU2_CNN_2D_41979010351328
// MI455X (gfx1250) — compile-verified
//
#include <hip/hip_runtime.h>
#include <math.h>

#define LSZ 512
#define NPLANE (LSZ * LSZ)   // 262144 = 2^18

// Flip to 0 if the async-LDS inline asm is rejected by the assembler.
#define USE_ASYNC_STAGE 1

typedef float v2f __attribute__((ext_vector_type(2)));
typedef float v8f __attribute__((ext_vector_type(8)));

// ---------------------------------------------------------------------------
// 16-byte global -> LDS copy. Async path: CDNA5 global_load_async_to_lds_b128
// (LDS byte offset in VDST VGPR, 64-bit global address in VADDR pair),
// completion tracked by ASYNCcnt.
// ---------------------------------------------------------------------------
__device__ __forceinline__ void stage16(float* lds, const float* g) {
#if USE_ASYNC_STAGE
  asm volatile("global_load_async_to_lds_b128 %0, %1, off"
               :: "v"((unsigned)(size_t)lds),
                  "v"((unsigned long long)(size_t)g)
               : "memory");
#else
  *(float4*)lds = *(const float4*)g;
#endif
}

__device__ __forceinline__ void stage_wait() {
#if USE_ASYNC_STAGE
#if __has_builtin(__builtin_amdgcn_s_wait_asynccnt)
  __builtin_amdgcn_s_wait_asynccnt(0);
#else
  asm volatile("s_wait_asynccnt 0x0" ::: "memory");
#endif
#endif
  __syncthreads();
}

__device__ __forceinline__ v8f wmma4(v2f a, v2f b, v8f c) {
  return __builtin_amdgcn_wmma_f32_16x16x4_f32(false, a, false, b, (short)0, c,
                                               false, false);
}

// ---------------------------------------------------------------------------
// Fused complex GEMM on the f32 WMMA pipe (all matrices 512x512 row-major).
//   C = (Ar + i*sa*Ai) * (Br + i*sb*Bi),  CONJ <=> sa*sb == -1
//   Cre = alphaRe * (Ar*Br - sasb*Ai*Bi)
//   Cim = alphaIm * (Ar*Bi + sasb*Ai*Br)      (alphaIm carries the sb sign)
// MODE: 0 = full complex, 1 = B real (Bi ignored), 2 = write real part only.
// Block 128 threads / 4 waves; block tile 64x64; wave tile 32x32 (2x2 WMMA).
// Signs are compile-time, so the only per-fragment VALU cost is one xor-negate.
// ---------------------------------------------------------------------------
template <int MODE, bool CONJ>
__global__ __launch_bounds__(128) void cgemm512(
    float* __restrict__ Cre, float* __restrict__ Cim,
    const float* __restrict__ Ar, const float* __restrict__ Ai,
    const float* __restrict__ Br, const float* __restrict__ Bi,
    float alphaRe, float alphaIm) {
  constexpr int BM = 64, BN = 64, BK = 16;
  __shared__ float sAr[BM][BK + 4];                      // [m][k], ds_load_b64 frags
  __shared__ float sAi[BM][BK + 4];
  __shared__ float sBr[BK][BN + 4];                      // [k][n]
  __shared__ float sBi[(MODE != 1) ? BK : 1][BN + 4];

  const int tid  = threadIdx.x;
  const int lane = tid & 31;
  const int wave = tid >> 5;
  const int wm   = (wave >> 1) * 32;
  const int wn   = (wave & 1) * 32;
  const int bm   = blockIdx.y * BM;
  const int bn   = blockIdx.x * BN;
  const int lh   = lane & 15;          // M (A frag) / N (B frag)
  const int kq   = (lane >> 4) << 1;   // half-wave K pair: 0 or 2

  v8f accRe[2][2] = {};
  v8f accIm[2][2] = {};                // dead if MODE==2

  for (int k0 = 0; k0 < LSZ; k0 += BK) {
    // A tiles: rows [bm,bm+64) x cols [k0,k0+16), contiguous 16B chunks
    for (int t = tid; t < BM * (BK / 4); t += 128) {
      const int m = t >> 2, q = (t & 3) << 2;
      stage16(&sAr[m][q], Ar + (bm + m) * LSZ + k0 + q);
      stage16(&sAi[m][q], Ai + (bm + m) * LSZ + k0 + q);
    }
    // B tiles: rows [k0,k0+16) x cols [bn,bn+64)
    for (int t = tid; t < BK * (BN / 4); t += 128) {
      const int kk = t >> 4, q = (t & 15) << 2;
      stage16(&sBr[kk][q], Br + (k0 + kk) * LSZ + bn + q);
      if (MODE != 1)
        stage16(&sBi[kk][q], Bi + (k0 + kk) * LSZ + bn + q);
    }
    stage_wait();

#pragma unroll
    for (int kk = 0; kk < BK; kk += 4) {
      v2f ar[2], u[2], nu[2], br[2], bi[2];
#pragma unroll
      for (int t = 0; t < 2; ++t) {
        const int m = wm + t * 16 + lh;
        ar[t] = *(const v2f*)&sAr[m][kk + kq];
        const v2f ai = *(const v2f*)&sAi[m][kk + kq];
        u[t]  = CONJ ? -ai : ai;   // +sasb*Ai  (feeds Cim chain)
        nu[t] = CONJ ? ai : -ai;   // -sasb*Ai  (feeds Cre chain)
        const int n = wn + t * 16 + lh;
        br[t].x = sBr[kk + kq + 0][n];
        br[t].y = sBr[kk + kq + 1][n];
        if (MODE != 1) {
          bi[t].x = sBi[kk + kq + 0][n];
          bi[t].y = sBi[kk + kq + 1][n];
        }
      }
#pragma unroll
      for (int i = 0; i < 2; ++i)
#pragma unroll
        for (int j = 0; j < 2; ++j) {
          accRe[i][j] = wmma4(ar[i], br[j], accRe[i][j]);
          if (MODE != 1)
            accRe[i][j] = wmma4(nu[i], bi[j], accRe[i][j]);
          if (MODE != 2) {
            if (MODE != 1)
              accIm[i][j] = wmma4(ar[i], bi[j], accIm[i][j]);
            accIm[i][j] = wmma4(u[i], br[j], accIm[i][j]);
          }
        }
    }
    __syncthreads();
  }

  // C/D layout: VGPR v -> row v (lanes 0-15) / row v+8 (lanes 16-31)
  const int rq = (lane >> 4) * 8;
#pragma unroll
  for (int i = 0; i < 2; ++i)
#pragma unroll
    for (int j = 0; j < 2; ++j) {
      const int col = bn + wn + j * 16 + lh;
#pragma unroll
      for (int v = 0; v < 8; ++v) {
        const int row = bm + wm + i * 16 + rq + v;
        Cre[row * LSZ + col] = alphaRe * accRe[i][j][v];
        if (MODE != 2)
          Cim[row * LSZ + col] = alphaIm * accIm[i][j][v];
      }
    }
}

// ---------------------------------------------------------------------------
// DFT matrix: F[m][n] = exp(-2*pi*i*m*n/L), mn reduced mod L.
// ---------------------------------------------------------------------------
__global__ void init_dft(float* __restrict__ Fr, float* __restrict__ Fi) {
  const int idx = blockIdx.x * blockDim.x + threadIdx.x;
  const int m = idx >> 9, n = idx & (LSZ - 1);
  const int t = (m * n) & (LSZ - 1);
  const float ang = -6.28318530717958647692f * (float)t / (float)LSZ;
  Fr[idx] = cosf(ang);
  Fi[idx] = sinf(ang);
}

// dihedral symmetry copy: bit0 = flip j, bit1 = flip i, bit2 = transpose
__global__ void make_sym(float* __restrict__ dst, const float* __restrict__ x,
                         int c, int s) {
  const int idx = blockIdx.x * blockDim.x + threadIdx.x;
  const int i = idx >> 9, j = idx & (LSZ - 1);
  const int ii = (s & 2) ? (LSZ - 1 - i) : i;
  const int jj = (s & 1) ? (LSZ - 1 - j) : j;
  const float* xc = x + (size_t)c * NPLANE;
  dst[idx] = (s & 4) ? xc[jj * LSZ + ii] : xc[ii * LSZ + jj];
}

// Y = conj(Kf) * Xf  (elementwise complex)
__global__ void specmul(float* __restrict__ Yre, float* __restrict__ Yim,
                        const float* __restrict__ Xre, const float* __restrict__ Xim,
                        const float* __restrict__ Kre, const float* __restrict__ Kim) {
  const int idx = blockIdx.x * blockDim.x + threadIdx.x;
  const float kr = Kre[idx], ki = Kim[idx];
  const float xr = Xre[idx], xi = Xim[idx];
  Yre[idx] = kr * xr + ki * xi;
  Yim[idx] = kr * xi - ki * xr;
}

// y0[i,j] = sum_c xs_c[i,j] * corr_c[i,j]
__global__ void input_layer(float* __restrict__ y0, const float* __restrict__ xs,
                            const float* __restrict__ corr) {
  const int idx = blockIdx.x * blockDim.x + threadIdx.x;
  float s = 0.f;
#pragma unroll
  for (int c = 0; c < 3; ++c)
    s += xs[c * NPLANE + idx] * corr[c * NPLANE + idx];
  y0[idx] = s;
}

// out[i,j,f] = elu( sum_{a,b,c} k[a,b,c,f] * in_c[(a-i)%L,(b-j)%L] )
__global__ void conv5_elu(float* __restrict__ out, const float* __restrict__ in,
                          const float* __restrict__ k, int Cin, int Fout) {
  const int idx = blockIdx.x * blockDim.x + threadIdx.x;
  const int f = idx >> 18;            // / NPLANE
  const int p = idx & (NPLANE - 1);
  const int i = p >> 9, j = p & (LSZ - 1);
  float s = 0.f;
  for (int a = 0; a < 5; ++a) {
    const int ri = (a - i) & (LSZ - 1);
    for (int b = 0; b < 5; ++b) {
      const int rj = (b - j) & (LSZ - 1);
      const float* kp = k + (size_t)((a * 5 + b) * Cin) * Fout + f;
      const float* ip = in + ri * LSZ + rj;
      for (int c = 0; c < Cin; ++c)
        s += kp[(size_t)c * Fout] * ip[(size_t)c * NPLANE];
    }
  }
  out[(size_t)f * NPLANE + p] = (s > 0.f) ? s : (expf(s) - 1.f);
}

__global__ void zero_acc(float* acc) {
  acc[0] = 0.f; acc[1] = 0.f; acc[2] = 0.f;
}

__global__ void reduce_sum(const float* __restrict__ src, int n, float* acc) {
  __shared__ float sh[256];
  float v = 0.f;
  for (int i = blockIdx.x * blockDim.x + threadIdx.x; i < n;
       i += gridDim.x * blockDim.x)
    v += src[i];
  sh[threadIdx.x] = v;
  __syncthreads();
  for (int o = 128; o > 0; o >>= 1) {
    if ((int)threadIdx.x < o) sh[threadIdx.x] += sh[threadIdx.x + o];
    __syncthreads();
  }
  if (threadIdx.x == 0) atomicAdd(acc, sh[0]);
}

__global__ void regu_reduce(const float* __restrict__ x,
                            const float* __restrict__ rlen, float* acc) {
  __shared__ float s1[256], s2[256];
  const float rl = rlen[0];
  const float inv = 1.f / (2.f * rl * rl);
  float a = 0.f, b = 0.f;
  for (int i = blockIdx.x * blockDim.x + threadIdx.x; i < NPLANE;
       i += gridDim.x * blockDim.x) {
    const float x0 = x[i], x1 = x[NPLANE + i], x2c = x[2 * NPLANE + i];
    const float q = x0 * x0 + x1 * x1 + x2c * x2c;
    a += q;
    b += expf(q * inv);
  }
  s1[threadIdx.x] = a; s2[threadIdx.x] = b;
  __syncthreads();
  for (int o = 128; o > 0; o >>= 1) {
    if ((int)threadIdx.x < o) {
      s1[threadIdx.x] += s1[threadIdx.x + o];
      s2[threadIdx.x] += s2[threadIdx.x + o];
    }
    __syncthreads();
  }
  if (threadIdx.x == 0) { atomicAdd(acc + 1, s1[0]); atomicAdd(acc + 2, s2[0]); }
}

__global__ void finalize_k(float* out, const float* acc,
                           const float* regu2, const float* regu) {
  out[0] = acc[0] * 0.125f - regu2[0] * acc[1] - regu[0] * acc[2];
}

// ---------------------------------------------------------------------------
// Orchestration. Workspace (floats): Fr Fi Kfr Kfi Tre Tim Xre Xim Yre Yim
// xs[3] corr[3] y0 y1[8] y2[8] acc[3]  => ~34.6 MB, L2-resident.
// ---------------------------------------------------------------------------
extern "C" void kernel_launch(void* const* d_in, const int* in_sizes, int n_in,
                              void* d_out, int out_size, void* d_ws, size_t ws_size,
                              hipStream_t stream) {
  (void)in_sizes; (void)n_in; (void)out_size; (void)ws_size;
  const float* x        = (const float*)d_in[0];   // (3,512,512)
  const float* Kin      = (const float*)d_in[1];   // (512,512)
  const float* k0       = (const float*)d_in[2];   // (5,5,1,8)
  const float* k1       = (const float*)d_in[3];   // (5,5,8,8)
  const float* regu2    = (const float*)d_in[4];
  const float* regu     = (const float*)d_in[5];
  const float* regu_len = (const float*)d_in[6];
  float* out = (float*)d_out;

  float* W    = (float*)d_ws;
  float* Fr   = W + (size_t)0  * NPLANE;
  float* Fi   = W + (size_t)1  * NPLANE;
  float* Kfr  = W + (size_t)2  * NPLANE;
  float* Kfi  = W + (size_t)3  * NPLANE;
  float* Tre  = W + (size_t)4  * NPLANE;
  float* Tim  = W + (size_t)5  * NPLANE;
  float* Xre  = W + (size_t)6  * NPLANE;
  float* Xim  = W + (size_t)7  * NPLANE;
  float* Yre  = W + (size_t)8  * NPLANE;
  float* Yim  = W + (size_t)9  * NPLANE;
  float* xs   = W + (size_t)10 * NPLANE;   // 3 planes
  float* corr = W + (size_t)13 * NPLANE;   // 3 planes
  float* y0   = W + (size_t)16 * NPLANE;   // 1 plane
  float* y1   = W + (size_t)17 * NPLANE;   // 8 planes
  float* y2   = W + (size_t)25 * NPLANE;   // 8 planes
  float* acc  = W + (size_t)33 * NPLANE;   // 3 scalars

  const dim3 gG(LSZ / 64, LSZ / 64), bG(128);
  // full complex, sa*sb = +1
  auto CGF = [&](float* cre, float* cim, const float* ar, const float* ai,
                 const float* br, const float* bi) {
    cgemm512<0, false><<<gG, bG, 0, stream>>>(cre, cim, ar, ai, br, bi, 1.f, 1.f);
  };
  // full complex, A conjugated (sa = -1, sb = +1)
  auto CGFc = [&](float* cre, float* cim, const float* ar, const float* ai,
                  const float* br, const float* bi) {
    cgemm512<0, true><<<gG, bG, 0, stream>>>(cre, cim, ar, ai, br, bi, 1.f, 1.f);
  };
  // B real: Cre = Ar*B, Cim = Ai*B
  auto CGB = [&](float* cre, float* cim, const float* ar, const float* ai,
                 const float* br) {
    cgemm512<1, false><<<gG, bG, 0, stream>>>(cre, cim, ar, ai, br, nullptr, 1.f, 1.f);
  };
  // real part only, B conjugated (sa = +1, sb = -1 => sasb = -1)
  auto CGRc = [&](float* cre, const float* ar, const float* ai,
                  const float* br, const float* bi, float al) {
    cgemm512<2, true><<<gG, bG, 0, stream>>>(cre, nullptr, ar, ai, br, bi, al, 0.f);
  };

  const int EB = 256, EG = NPLANE / EB;

  zero_acc<<<1, 1, 0, stream>>>(acc);
  init_dft<<<EG, EB, 0, stream>>>(Fr, Fi);

  // Kf = F * K * F   (F symmetric, K real)
  CGB(Tre, Tim, Fr, Fi, Kin);
  CGF(Kfr, Kfi, Tre, Tim, Fr, Fi);

  const float invN = 1.f / (float)NPLANE;
  for (int s = 0; s < 8; ++s) {
    for (int c = 0; c < 3; ++c) {
      float* xc = xs + (size_t)c * NPLANE;
      float* cc = corr + (size_t)c * NPLANE;
      make_sym<<<EG, EB, 0, stream>>>(xc, x, c, s);
      CGB(Tre, Tim, Fr, Fi, xc);                          // G  = F * x
      CGF(Xre, Xim, Tre, Tim, Fr, Fi);                    // Xf = G * F
      specmul<<<EG, EB, 0, stream>>>(Yre, Yim, Xre, Xim, Kfr, Kfi);
      CGFc(Tre, Tim, Fr, Fi, Yre, Yim);                   // T2 = conj(F) * Y
      CGRc(cc, Tre, Tim, Fr, Fi, invN);                   // corr = Re(T2*conj(F))/N
    }
    input_layer<<<EG, EB, 0, stream>>>(y0, xs, corr);
    conv5_elu<<<8 * (NPLANE / EB), EB, 0, stream>>>(y1, y0, k0, 1, 8);
    conv5_elu<<<8 * (NPLANE / EB), EB, 0, stream>>>(y2, y1, k1, 8, 8);
    reduce_sum<<<512, 256, 0, stream>>>(y2, 8 * NPLANE, acc);
  }
  regu_reduce<<<512, 256, 0, stream>>>(x, regu_len, acc);
  finalize_k<<<1, 1, 0, stream>>>(out, acc, regu2, regu);
}